// GCN_59244778881669
// MI455X (gfx1250) — compile-verified
//
#include <hip/hip_runtime.h>
#include <hip/hip_bf16.h>

#define F_IN 512
#define HID  16
#define NCLS 7
#define TILES_PER_WAVE 2

typedef float v2f __attribute__((ext_vector_type(2)));
typedef float v8f __attribute__((ext_vector_type(8)));

// ---------------- degree / norm prep ----------------

__global__ void gcn_init_deg(float* __restrict__ deg, int n) {
    int i = blockIdx.x * blockDim.x + threadIdx.x;
    if (i < n) deg[i] = 1.0f;   // self-loop contributes 1 to every node's degree
}

__global__ void gcn_deg_scatter(const int* __restrict__ col, float* __restrict__ deg, int e) {
    int i = blockIdx.x * blockDim.x + threadIdx.x;
    if (i < e) atomicAdd(&deg[col[i]], 1.0f);
}

__global__ void gcn_dinv(float* __restrict__ degdinv, int n) {
    int i = blockIdx.x * blockDim.x + threadIdx.x;
    if (i < n) degdinv[i] = rsqrtf(degdinv[i]);   // deg >= 1 always (self-loops)
}

// ---------------- conv1 GEMM: h1 = x @ W1 via V_WMMA_F32_16X16X4_F32 ----------------
// One wave computes TILES_PER_WAVE 16x16 output tiles sharing the same B (W1)
// fragment each K-step, accumulating over K=512 in steps of 4.

__device__ __forceinline__ void store_tile(float* __restrict__ h1, const v8f& acc,
                                           int tile, int lane, int n) {
    // D layout: VGPR v -> row tile*16 + v (+8 for upper half-wave), col = lane&15
    int mbase = tile * 16 + ((lane >> 4) << 3);
    int ncol  = lane & 15;
    float* p = h1 + (size_t)mbase * HID + ncol;
    if (mbase + 8 <= n) {              // full tile in range: unguarded store clause
        #pragma unroll
        for (int v = 0; v < 8; ++v) p[v * HID] = acc[v];
    } else {
        #pragma unroll
        for (int v = 0; v < 8; ++v) if (mbase + v < n) p[v * HID] = acc[v];
    }
}

__global__ void gcn_gemm1_wmma(const float* __restrict__ x, const float* __restrict__ W1,
                               float* __restrict__ h1, int n) {
    int lane = threadIdx.x & 31;
    int wid  = (blockIdx.x * blockDim.x + threadIdx.x) >> 5;
    int ntiles = (n + 15) >> 4;
    int tile0 = wid * TILES_PER_WAVE;
    if (tile0 >= ntiles) return;
    bool has1 = (tile0 + 1) < ntiles;

    int m    = lane & 15;            // row within tile (A) / col within tile (B)
    int koff = (lane >> 4) << 1;     // lanes 16-31 hold K+2, K+3

    int row0 = tile0 * 16 + m;
    int row1 = row0 + 16;
    const float* xa0 = x + (size_t)(row0 < n ? row0 : 0) * F_IN + koff;
    const float* xa1 = x + (size_t)(row1 < n ? row1 : 0) * F_IN + koff;

    v8f acc0 = {}, acc1 = {};
    #pragma unroll 4
    for (int k = 0; k < F_IN; k += 4) {
        v2f b;
        b.x = W1[(size_t)(k + koff)     * HID + m];
        b.y = W1[(size_t)(k + koff + 1) * HID + m];
        v2f a0 = *(const v2f*)(xa0 + k);   // 8B-aligned: koff is even
        v2f a1 = *(const v2f*)(xa1 + k);
        // (neg_a, A, neg_b, B, c_mod, C, reuse_a, reuse_b)
        acc0 = __builtin_amdgcn_wmma_f32_16x16x4_f32(
            false, a0, false, b, (short)0, acc0, false, false);
        acc1 = __builtin_amdgcn_wmma_f32_16x16x4_f32(
            false, a1, false, b, (short)0, acc1, false, false);
    }

    store_tile(h1, acc0, tile0, lane, n);
    if (has1) store_tile(h1, acc1, tile0 + 1, lane, n);
}

// ---------------- conv1 aggregation ----------------

__global__ void gcn_init_out1(const float* __restrict__ h1, const float* __restrict__ dinv,
                              const float* __restrict__ b1, float* __restrict__ out1, int n) {
    int i = blockIdx.x * blockDim.x + threadIdx.x;
    if (i >= n) return;
    float s = dinv[i]; s = s * s;                    // self-loop norm = dinv[i]^2
    const float4* hp = (const float4*)(h1 + (size_t)i * HID);
    float4* op = (float4*)(out1 + (size_t)i * HID);
    #pragma unroll
    for (int q = 0; q < 4; ++q) {
        float4 h = hp[q];
        op[q] = make_float4(b1[4*q+0] + s*h.x, b1[4*q+1] + s*h.y,
                            b1[4*q+2] + s*h.z, b1[4*q+3] + s*h.w);
    }
}

__global__ void gcn_scatter1(const int* __restrict__ rowi, const int* __restrict__ coli,
                             const float* __restrict__ dinv, const float* __restrict__ h1,
                             float* __restrict__ out1, int e) {
    int idx = blockIdx.x * blockDim.x + threadIdx.x;
    if (idx >= e) return;
    int r = rowi[idx], c = coli[idx];
    float w = dinv[r] * dinv[c];
    const float* hp = h1 + (size_t)r * HID;      // 6.4 MB, L2-resident gather
    float* op = out1 + (size_t)c * HID;
    #pragma unroll
    for (int q = 0; q < HID; ++q) atomicAdd(op + q, w * hp[q]);
}

// ---------------- conv2: ReLU + tiny GEMV + self-loop init ----------------

__global__ void gcn_gemm2(const float* __restrict__ out1, const float* __restrict__ W2,
                          const float* __restrict__ b2, const float* __restrict__ dinv,
                          float* __restrict__ g, float* __restrict__ out2, int n) {
    int i = blockIdx.x * blockDim.x + threadIdx.x;
    if (i >= n) return;
    float h[HID];
    const float* ip = out1 + (size_t)i * HID;
    #pragma unroll
    for (int c = 0; c < HID; ++c) h[c] = fmaxf(ip[c], 0.0f);   // ReLU
    float s = dinv[i]; s = s * s;
    #pragma unroll
    for (int c2 = 0; c2 < NCLS; ++c2) {
        float acc = 0.0f;
        #pragma unroll
        for (int c = 0; c < HID; ++c) acc += h[c] * W2[c * NCLS + c2];
        g[(size_t)i * NCLS + c2]    = acc;
        out2[(size_t)i * NCLS + c2] = b2[c2] + s * acc;
    }
}

__global__ void gcn_scatter2(const int* __restrict__ rowi, const int* __restrict__ coli,
                             const float* __restrict__ dinv, const float* __restrict__ g,
                             float* __restrict__ out2, int e) {
    int idx = blockIdx.x * blockDim.x + threadIdx.x;
    if (idx >= e) return;
    int r = rowi[idx], c = coli[idx];
    float w = dinv[r] * dinv[c];
    const float* gp = g + (size_t)r * NCLS;
    float* op = out2 + (size_t)c * NCLS;
    #pragma unroll
    for (int q = 0; q < NCLS; ++q) atomicAdd(op + q, w * gp[q]);
}

// ---------------- log_softmax ----------------

__global__ void gcn_logsoftmax(const float* __restrict__ out2, float* __restrict__ out, int n) {
    int i = blockIdx.x * blockDim.x + threadIdx.x;
    if (i >= n) return;
    float v[NCLS];
    float m = -1e30f;
    #pragma unroll
    for (int c = 0; c < NCLS; ++c) { v[c] = out2[(size_t)i * NCLS + c]; m = fmaxf(m, v[c]); }
    float sum = 0.0f;
    #pragma unroll
    for (int c = 0; c < NCLS; ++c) sum += __expf(v[c] - m);
    float lse = m + __logf(sum);
    #pragma unroll
    for (int c = 0; c < NCLS; ++c) out[(size_t)i * NCLS + c] = v[c] - lse;
}

// ---------------- launcher ----------------

extern "C" void kernel_launch(void* const* d_in, const int* in_sizes, int n_in,
                              void* d_out, int out_size, void* d_ws, size_t ws_size,
                              hipStream_t stream) {
    const float* x  = (const float*)d_in[0];
    const int*   ei = (const int*)  d_in[1];   // integer inputs arrive as int32
    const float* W1 = (const float*)d_in[2];
    const float* b1 = (const float*)d_in[3];
    const float* W2 = (const float*)d_in[4];
    const float* b2 = (const float*)d_in[5];
    float* out = (float*)d_out;

    int n = in_sizes[0] / F_IN;
    int e = in_sizes[1] / 2;
    const int* rowi = ei;        // edge_index[0] = sources
    const int* coli = ei + e;    // edge_index[1] = targets

    // workspace layout (floats): dinv[n] | h1[n*16] | out1[n*16] | g[n*7] | out2[n*7]
    float* ws   = (float*)d_ws;
    float* dinv = ws;
    float* h1   = dinv + n;
    float* out1 = h1   + (size_t)n * HID;
    float* g    = out1 + (size_t)n * HID;
    float* out2 = g    + (size_t)n * NCLS;

    const int B = 256;
    int gn = (n + B - 1) / B;
    int ge = (e + B - 1) / B;

    gcn_init_deg   <<<gn, B, 0, stream>>>(dinv, n);
    gcn_deg_scatter<<<ge, B, 0, stream>>>(coli, dinv, e);
    gcn_dinv       <<<gn, B, 0, stream>>>(dinv, n);

    int ntiles = (n + 15) >> 4;
    int nwaves = (ntiles + TILES_PER_WAVE - 1) / TILES_PER_WAVE;
    int gw = (nwaves * 32 + B - 1) / B;
    gcn_gemm1_wmma <<<gw, B, 0, stream>>>(x, W1, h1, n);

    gcn_init_out1  <<<gn, B, 0, stream>>>(h1, dinv, b1, out1, n);
    gcn_scatter1   <<<ge, B, 0, stream>>>(rowi, coli, dinv, h1, out1, e);

    gcn_gemm2      <<<gn, B, 0, stream>>>(out1, W2, b2, dinv, g, out2, n);
    gcn_scatter2   <<<ge, B, 0, stream>>>(rowi, coli, dinv, g, out2, e);

    gcn_logsoftmax <<<gn, B, 0, stream>>>(out2, out, n);
}